// spacial_attention_37615323578990
// MI455X (gfx1250) — compile-verified
//
#include <hip/hip_runtime.h>
#include <hip/hip_bf16.h>

// ---------------------------------------------------------------------------
// Spatial attention (CBAM-style) for MI455X / gfx1250.
//   X:[32,256,64,64] f32 -> channel max/avg pool -> 7x7 conv(2->1) -> sigmoid
//   -> gate * X.
// Memory-bound: ~268MB min HBM traffic -> ~11.5us @ 23.3TB/s.
//   K1a/K1b: channel reduction, 4-way channel split for 4096 waves of MLP
//            (needs ~16MB in flight to cover HBM latency at 23.3TB/s).
//   K2:      tiny 98-tap conv as chained V_WMMA_F32_16X16X4_F32.
//   K3:      NT streaming gate*X (X ideally still L2-resident: 128MB < 192MB).
// ---------------------------------------------------------------------------

typedef float v2f __attribute__((ext_vector_type(2)));
typedef float v4f __attribute__((ext_vector_type(4)));
typedef float v8f __attribute__((ext_vector_type(8)));

#define BATCH 32
#define CHN   256
#define HWSZ  4096   // 64*64
#define HWG4  1024   // HWSZ/4
#define NPART 4      // channel-split factor
#define CPART (CHN / NPART)          // 64 channels per partial
#define NPGRP (BATCH * HWG4)         // 32768 float4 pixel-groups total

// ---------------------------------------------------------------------------
// K1a: partial channel max & sum. 131072 threads; thread t = (part, pixgrp).
// Each reduces 64 channels with float4 loads; unroll 16 keeps ~16 loads
// outstanding per wave for latency hiding. RT loads keep X hot in L2 for K3.
// ---------------------------------------------------------------------------
__global__ __launch_bounds__(256) void pool_partial_kernel(
    const float* __restrict__ X,
    float* __restrict__ part_max,    // [NPART][NPGRP] v4f
    float* __restrict__ part_sum)    // [NPART][NPGRP] v4f
{
    int t    = blockIdx.x * 256 + threadIdx.x;   // 0 .. 131071
    int part = t >> 15;                          // 0..3
    int pg   = t & (NPGRP - 1);                  // pixel-group id
    int b    = pg >> 10;
    int g    = pg & (HWG4 - 1);

    const v4f* X4 = (const v4f*)X;
    size_t base = (size_t)b * (CHN * HWG4) + (size_t)(part * CPART) * HWG4 + g;

    v4f v  = X4[base];
    v4f mx = v;
    v4f sm = v;
    #pragma unroll 16
    for (int c = 1; c < CPART; ++c) {
        v4f x = X4[base + (size_t)c * HWG4];
        mx.x = fmaxf(mx.x, x.x); mx.y = fmaxf(mx.y, x.y);
        mx.z = fmaxf(mx.z, x.z); mx.w = fmaxf(mx.w, x.w);
        sm += x;
    }
    ((v4f*)part_max)[t] = mx;
    ((v4f*)part_sum)[t] = sm;
}

// ---------------------------------------------------------------------------
// K1b: combine the 4 partials -> final max / avg maps (1MB, L2-resident).
// ---------------------------------------------------------------------------
__global__ __launch_bounds__(256) void pool_combine_kernel(
    const float* __restrict__ part_max,
    const float* __restrict__ part_sum,
    float* __restrict__ pmax,
    float* __restrict__ pavg)
{
    int t = blockIdx.x * 256 + threadIdx.x;      // 0 .. 32767 (pixel-group)
    const v4f* PM = (const v4f*)part_max;
    const v4f* PS = (const v4f*)part_sum;

    v4f mx = PM[t];
    v4f sm = PS[t];
    #pragma unroll
    for (int p = 1; p < NPART; ++p) {
        v4f m = PM[p * NPGRP + t];
        v4f s = PS[p * NPGRP + t];
        mx.x = fmaxf(mx.x, m.x); mx.y = fmaxf(mx.y, m.y);
        mx.z = fmaxf(mx.z, m.z); mx.w = fmaxf(mx.w, m.w);
        sm += s;
    }
    ((v4f*)pmax)[t] = mx;
    ((v4f*)pavg)[t] = sm * (1.0f / (float)CHN);
}

// ---------------------------------------------------------------------------
// K2: 7x7 conv (2ch -> 1) + bias + sigmoid via V_WMMA_F32_16X16X4_F32.
// Wave handles 16 pixels. A(16x4) = weights w[k0..k0+3] broadcast to all 16
// rows (A layout: lanes 0-15 hold K=0,1; lanes 16-31 hold K=2,3). B(4x16) =
// im2col taps for the wave's 16 pixels (lane n<16: pixel n taps k0,k0+1;
// lane n>=16: pixel n-16 taps k0+2,k0+3). All rows of D are then identical,
// so each lane's acc[0] is the conv result for its own pixel (16x16 f32 C/D
// layout: VGPR0 lane n -> D[0][n] / D[8][n-16]). 25 chained WMMAs cover
// 98 taps padded to 100 with zero weights. EXEC is all-1s at each WMMA
// (tap bound checks reconverge before the matrix op).
// ---------------------------------------------------------------------------
__device__ __forceinline__ float conv_tap(
    const float* __restrict__ pmax, const float* __restrict__ pavg,
    int b, int h, int w, int kk)
{
    if (kk >= 98) return 0.0f;                     // zero-padded taps
    int ch = (kk >= 49) ? 1 : 0;                   // 0 = max-pool, 1 = avg-pool
    int r  = kk - ch * 49;
    int ky = r / 7;
    int kx = r - ky * 7;
    int ih = h + ky - 3;
    int iw = w + kx - 3;
    if ((unsigned)ih >= 64u || (unsigned)iw >= 64u) return 0.0f;  // SAME pad
    const float* p = ch ? pavg : pmax;
    return p[(b << 12) + (ih << 6) + iw];
}

__global__ __launch_bounds__(256) void conv_gate_kernel(
    const float* __restrict__ pmax,
    const float* __restrict__ pavg,
    const float* __restrict__ cw,    // [1,2,7,7] = 98 floats
    const float* __restrict__ cb,    // [1]
    float* __restrict__ gate)        // [32,4096]
{
    __shared__ float wsm[100];
    if (threadIdx.x < 100)
        wsm[threadIdx.x] = (threadIdx.x < 98) ? cw[threadIdx.x] : 0.0f;
    __syncthreads();

    int lane = threadIdx.x & 31;
    int gw   = blockIdx.x * 8 + (threadIdx.x >> 5);  // global wave id
    int pix  = gw * 16 + (lane & 15);                // this lane's pixel
    int b    = pix >> 12;
    int hw   = pix & 4095;
    int h    = hw >> 6;
    int w    = hw & 63;
    int khalf = (lane >> 4) << 1;    // 0 for lanes 0-15, 2 for lanes 16-31

    v8f acc = {0.f, 0.f, 0.f, 0.f, 0.f, 0.f, 0.f, 0.f};
    for (int k0 = 0; k0 < 100; k0 += 4) {
        v2f a, bb;
        a.x = wsm[k0 + khalf + 0];
        a.y = wsm[k0 + khalf + 1];
        int kk = k0 + khalf;
        bb.x = conv_tap(pmax, pavg, b, h, w, kk + 0);
        bb.y = conv_tap(pmax, pavg, b, h, w, kk + 1);
        // D = A x B + C  (8 args: neg_a, A, neg_b, B, c_mod, C, reuse_a, reuse_b)
        acc = __builtin_amdgcn_wmma_f32_16x16x4_f32(
            false, a, false, bb, (short)0, acc, false, false);
    }

    if (lane < 16) {
        float x = acc[0] + cb[0];
        gate[pix] = 1.0f / (1.0f + expf(-x));
    }
}

// ---------------------------------------------------------------------------
// K3: out = gate * X, float4 streaming. X loads NT (last use, served from L2
// lines left resident by K1a), stores NT so the 134MB write stream doesn't
// evict useful L2 lines mid-kernel.
// ---------------------------------------------------------------------------
__global__ __launch_bounds__(256) void apply_gate_kernel(
    const float* __restrict__ X,
    const float* __restrict__ gate,
    float* __restrict__ out)
{
    size_t t = (size_t)blockIdx.x * 256 + threadIdx.x;  // float4 index
    int hwg = (int)(t & (HWG4 - 1));
    int bc  = (int)(t >> 10);
    int b   = bc >> 8;

    v4f x = __builtin_nontemporal_load((const v4f*)X + t);
    v4f g = ((const v4f*)gate)[(b << 10) + hwg];
    v4f o = x * g;
    __builtin_nontemporal_store(o, (v4f*)out + t);
}

// ---------------------------------------------------------------------------
extern "C" void kernel_launch(void* const* d_in, const int* in_sizes, int n_in,
                              void* d_out, int out_size, void* d_ws, size_t ws_size,
                              hipStream_t stream)
{
    const float* X  = (const float*)d_in[0];   // [32,256,64,64]
    const float* cw = (const float*)d_in[1];   // [1,2,7,7]
    const float* cb = (const float*)d_in[2];   // [1]
    float* out = (float*)d_out;

    // Workspace layout (floats):
    float* part_max = (float*)d_ws;                       // 4*131072 = 524288
    float* part_sum = part_max + NPART * NPGRP * 4;       // 524288
    float* pmax     = part_sum + NPART * NPGRP * 4;       // 131072
    float* pavg     = pmax + BATCH * HWSZ;                // 131072
    float* gate     = pavg + BATCH * HWSZ;                // 131072  (~5.5MB total)

    // K1a: 4 * 32768 = 131072 threads (4096 waves)
    pool_partial_kernel<<<512, 256, 0, stream>>>(X, part_max, part_sum);
    // K1b: 32768 threads
    pool_combine_kernel<<<128, 256, 0, stream>>>(part_max, part_sum, pmax, pavg);
    // K2: 131072 pixels / 16 per wave = 8192 waves = 1024 blocks of 8 waves
    conv_gate_kernel<<<1024, 256, 0, stream>>>(pmax, pavg, cw, cb, gate);
    // K3: 33554432/4 = 8388608 threads
    apply_gate_kernel<<<32768, 256, 0, stream>>>(X, gate, out);
}